// LocalNetwork_82617990906159
// MI455X (gfx1250) — compile-verified
//
#include <hip/hip_runtime.h>

typedef float v2f __attribute__((ext_vector_type(2)));
typedef float v8f __attribute__((ext_vector_type(8)));

#define BSZ   256
#define DD    15
#define HH    63
#define WW    126
#define HW    (HH*WW)        // 7938
#define CELLS (DD*HW)        // 119070
#define OD    3
#define OH    13
#define OW    26
#define NPOS  (OD*OH*OW)     // 1014
#define KKP   128            // 125 reduction elems padded to 128

// workspace layout (float units)
#define OFF_YRAW  0                         // 256*1024
#define OFF_WL    (OFF_YRAW + BSZ*1024)     // 1014*128
#define OFF_VV    (OFF_WL   + NPOS*KKP)     // 1014*128
#define OFF_CIDX  (OFF_VV   + NPOS*KKP)     // 1014*128 (int)
#define OFF_S     (OFF_CIDX + NPOS*KKP)     // 1024
#define OFF_PMIN  (OFF_S    + 1024)         // 256*80
#define OFF_PMOUT (OFF_PMIN + BSZ*80)       // 256*80
#define OFF_MIN   (OFF_PMOUT+ BSZ*80)       // 256
#define OFF_MOUT  (OFF_MIN  + BSZ)          // 256

// ---------------------------------------------------------------------------
// Kernel 1: per-position tables. 1014 blocks x 128 threads.
//   WL[pos][kk]   = W[pos][kk] * land[cell]  (0 for pad/invalid/kk>=125)
//   VV[pos][kk]   = vol[cell]                (0 for pad/invalid)  -> mass_in
//   CIDX[pos][kk] = clamped flat cell index  (always safe to load)
//   S[pos]        = sum land*vol over valid cells -> mass_out weights
// ---------------------------------------------------------------------------
__global__ void ln_prep_kernel(const float* __restrict__ land,
                               const float* __restrict__ vol,
                               const float* __restrict__ Wt,
                               float* __restrict__ ws) {
    const int pos = blockIdx.x;
    const int kk  = threadIdx.x;
    const int i   = pos / (OH*OW);
    const int rem = pos - i*(OH*OW);
    const int j   = rem / OW;
    const int k   = rem - j*OW;

    const int kkc = min(kk, 124);
    const int p   = kkc / 25;
    const int q   = (kkc / 5) % 5;
    const int r   = kkc % 5;
    const int d   = 5*i + p;            // always < 15
    const int h   = 5*j - 1 + q;        // PH = 1
    const int w   = 5*k - 2 + r;        // PW = 2
    const int hc  = min(max(h, 0), HH-1);
    const int wc  = min(max(w, 0), WW-1);
    const int cell = d*HW + hc*WW + wc;
    const bool ok = (kk < 125) && (h >= 0) && (h < HH) && (w >= 0) && (w < WW);

    float lv = 0.f, vv = 0.f, wv = 0.f;
    if (ok) { lv = land[cell]; vv = vol[cell]; }
    if (kk < 125) wv = Wt[pos*125 + kk];

    ws[OFF_WL + pos*KKP + kk] = wv * lv;   // land folded into weight
    ws[OFF_VV + pos*KKP + kk] = vv;        // validity-masked volume
    ((int*)ws)[OFF_CIDX + pos*KKP + kk] = cell;

    __shared__ float red[128];
    red[kk] = lv * vv;
    __syncthreads();
    for (int s = 64; s > 0; s >>= 1) {
        if (kk < s) red[kk] += red[kk + s];
        __syncthreads();
    }
    if (kk == 0) ws[OFF_S + pos] = red[0];
}

// ---------------------------------------------------------------------------
// Kernel 2: WMMA block reduction. 1 wave per block.
// grid = (16 batch tiles, 39 (i,j) pairs, 2 half-rows of k)
// A[16x4]  = x for 16 batches, 4 reduction elems (lanes 0-15: K0,K1; 16-31: K2,K3)
// B[4x16]  = WL chunk broadcast to all 16 columns -> D[m,n] = dot for batch m.
// ---------------------------------------------------------------------------
__global__ void ln_wmma_kernel(const float* __restrict__ x,
                               const float* __restrict__ bias,
                               float* __restrict__ ws) {
    const int lane = threadIdx.x;       // 0..31
    const int sel  = lane >> 4;         // 0: K=0,1  1: K=2,3
    const int mrow = lane & 15;         // batch row of A held by this lane
    const int b0   = blockIdx.x * 16;
    const int ij   = blockIdx.y;        // 0..38
    const int i    = ij / OH;
    const int j    = ij - i*OH;
    const int kst  = blockIdx.z * (OW/2);
    const int gslot = blockIdx.y * 2 + blockIdx.z;   // 0..77 partial slot

    const float* __restrict__ WL = ws + OFF_WL;
    const float* __restrict__ VV = ws + OFF_VV;
    const int*   __restrict__ CI = ((const int*)ws) + OFF_CIDX;
    const float* __restrict__ Sv = ws + OFF_S;
    float* __restrict__ yraw  = ws + OFF_YRAW;
    float* __restrict__ pmin  = ws + OFF_PMIN;
    float* __restrict__ pmout = ws + OFF_PMOUT;

    const int xb = (b0 + mrow) * CELLS;

    float acc_mi = 0.f;                                   // x*vol partial
    v8f acc_mo = {0.f,0.f,0.f,0.f,0.f,0.f,0.f,0.f};       // relu(y)*S partial

    for (int kp = 0; kp < OW/2; ++kp) {
        const int k   = kst + kp;
        const int pos = (i*OH + j)*OW + k;
        const float* wlp = WL + pos*KKP;
        const float* vvp = VV + pos*KKP;
        const int*   cip = CI + pos*KKP;

        v8f c = {0.f,0.f,0.f,0.f,0.f,0.f,0.f,0.f};
        #pragma unroll
        for (int c32 = 0; c32 < 32; ++c32) {
            const int kb = 4*c32 + 2*sel;
            const int2  ci  = *(const int2*)(cip + kb);   // 8B aligned (kb even)
            const float x0  = x[xb + ci.x];
            const float x1  = x[xb + ci.y];
            const v2f   bm  = *(const v2f*)(wlp + kb);    // weight chunk (xlanes share)
            const v2f   vv2 = *(const v2f*)(vvp + kb);
            acc_mi = fmaf(x0, vv2.x, fmaf(x1, vv2.y, acc_mi));
            v2f a; a.x = x0; a.y = x1;
            c = __builtin_amdgcn_wmma_f32_16x16x4_f32(
                    false, a, false, bm, (short)0, c, false, false);
        }

        const float bb = bias[pos];
        const float Sp = Sv[pos];
        v8f yr;
        #pragma unroll
        for (int v = 0; v < 8; ++v) {
            float t = c[v] + bb;
            t = t > 0.f ? t : 0.f;                        // relu
            yr[v] = t;
            acc_mo[v] += t * Sp;
        }
        // C layout: lanes 0-15 vgpr v -> M=v ; lanes 16-31 -> M=v+8 (cols equal)
        if (lane == 0) {
            #pragma unroll
            for (int v = 0; v < 8; ++v) yraw[(b0 + v)*1024 + pos] = yr[v];
        } else if (lane == 16) {
            #pragma unroll
            for (int v = 0; v < 8; ++v) yraw[(b0 + 8 + v)*1024 + pos] = yr[v];
        }
    }

    // mass_in: lanes m and m+16 hold the same batch (different K slots)
    acc_mi += __shfl_xor(acc_mi, 16);
    if (lane < 16) pmin[(b0 + lane)*80 + gslot] = acc_mi;
    if (lane == 0) {
        #pragma unroll
        for (int v = 0; v < 8; ++v) pmout[(b0 + v)*80 + gslot] = acc_mo[v];
    } else if (lane == 16) {
        #pragma unroll
        for (int v = 0; v < 8; ++v) pmout[(b0 + 8 + v)*80 + gslot] = acc_mo[v];
    }
}

// ---------------------------------------------------------------------------
// Kernel 3: deterministic mass reduction, 1 block x 256 threads.
// ---------------------------------------------------------------------------
__global__ void ln_mass_kernel(float* __restrict__ ws) {
    const int b = threadIdx.x;
    float mi = 0.f, mo = 0.f;
    for (int g = 0; g < 78; ++g) {
        mi += ws[OFF_PMIN  + b*80 + g];
        mo += ws[OFF_PMOUT + b*80 + g];
    }
    ws[OFF_MIN  + b] = mi;
    ws[OFF_MOUT + b] = mo;
}

// ---------------------------------------------------------------------------
// Kernel 4: upsample/crop/mask/rescale — the 122 MB coalesced write.
// ---------------------------------------------------------------------------
__global__ void ln_out_kernel(const float* __restrict__ land,
                              const float* __restrict__ ws,
                              float* __restrict__ out) {
    const int b    = blockIdx.y;
    const int cell = blockIdx.x * blockDim.x + threadIdx.x;
    if (cell >= CELLS) return;
    const int w = cell % WW;
    const int t = cell / WW;
    const int h = t % HH;
    const int d = t / HH;
    const int pos = (d/5)*(OH*OW) + ((h+1)/5)*OW + ((w+2)/5);
    const float scale = ws[OFF_MIN + b] / ws[OFF_MOUT + b];
    out[b*CELLS + cell] = ws[OFF_YRAW + b*1024 + pos] * land[cell] * scale;
}

// ---------------------------------------------------------------------------
extern "C" void kernel_launch(void* const* d_in, const int* in_sizes, int n_in,
                              void* d_out, int out_size, void* d_ws, size_t ws_size,
                              hipStream_t stream) {
    (void)in_sizes; (void)n_in; (void)out_size; (void)ws_size;
    const float* x    = (const float*)d_in[0];
    const float* land = (const float*)d_in[1];
    const float* vol  = (const float*)d_in[2];
    const float* Wt   = (const float*)d_in[3];
    const float* bias = (const float*)d_in[4];
    float* ws  = (float*)d_ws;
    float* out = (float*)d_out;

    ln_prep_kernel<<<NPOS, 128, 0, stream>>>(land, vol, Wt, ws);

    dim3 g2(16, OD*OH, 2);
    ln_wmma_kernel<<<g2, 32, 0, stream>>>(x, bias, ws);

    ln_mass_kernel<<<1, 256, 0, stream>>>(ws);

    dim3 g3((CELLS + 255)/256, BSZ);
    ln_out_kernel<<<g3, 256, 0, stream>>>(land, ws, out);
}